// CSR_Attention_47991964565993
// MI455X (gfx1250) — compile-verified
//
#include <hip/hip_runtime.h>
#include <math.h>

#define EPSC 1e-8f
#define LN_EPS 1e-5f

typedef float v2f __attribute__((ext_vector_type(2)));
typedef float v8f __attribute__((ext_vector_type(8)));

static constexpr int B = 16, L = 512, D = 64, IN = 64, LP = 8;
static constexpr int CHUNKS = 8, CLEN = 64;         // L = CHUNKS * CLEN
static constexpr int ROWS = B * L;                  // 8192

// ---------------- reductions (wave32) ----------------
__device__ __forceinline__ void wave_red2(float& a, float& b) {
  for (int off = 16; off > 0; off >>= 1) {
    a += __shfl_down(a, off, 32);
    b += __shfl_down(b, off, 32);
  }
}

// all 256 threads must call; returns block-wide sums of (a,b)
__device__ __forceinline__ void block_red2_256(float a, float b, float* red,
                                               float& oa, float& ob) {
  int lane = threadIdx.x & 31, w = threadIdx.x >> 5;
  wave_red2(a, b);
  if (lane == 0) { red[w] = a; red[8 + w] = b; }
  __syncthreads();
  if (threadIdx.x == 0) {
    float s = 0.f, t = 0.f;
    for (int i = 0; i < 8; ++i) { s += red[i]; t += red[8 + i]; }
    red[0] = s; red[8] = t;
  }
  __syncthreads();
  oa = red[0]; ob = red[8];
  __syncthreads();
}

// ---------------- K1: projections via V_WMMA_F32_16X16X4_F32 ----------------
// grid (ROWS/16, 3), block 128 (4 waves, one 16-col N tile each)
__global__ void proj_wmma(const float* __restrict__ X,
                          const float* __restrict__ Wq, const float* __restrict__ bq,
                          const float* __restrict__ Wk, const float* __restrict__ bk,
                          const float* __restrict__ Wv, const float* __restrict__ bv,
                          float* __restrict__ Q, float* __restrict__ Zr,
                          float* __restrict__ V) {
  const int lane = threadIdx.x & 31;
  const int wvid = threadIdx.x >> 5;     // N tile 0..3
  const int half = lane >> 4;            // lane half selects K pair
  const int mn   = lane & 15;
  const int mrow = blockIdx.x * 16;
  const int which = blockIdx.y;
  const float* W; const float* bias;
  if (which == 0)      { W = Wq; bias = bq; }
  else if (which == 1) { W = Wk; bias = bk; }
  else                 { W = Wv; bias = bv; }
  const int ncol = wvid * 16 + mn;
  v8f acc = {};
  for (int kb = 0; kb < IN; kb += 4) {
    const int k0 = kb + half * 2;        // 16x4 f32 A: lanes<16 hold K0,K1; lanes>=16 K2,K3
    v2f a, bm;
    a.x  = X[(mrow + mn) * IN + k0];
    a.y  = X[(mrow + mn) * IN + k0 + 1];
    bm.x = W[(k0) * D + ncol];           // 4x16 B: row striped across lanes per VGPR
    bm.y = W[(k0 + 1) * D + ncol];
    acc = __builtin_amdgcn_wmma_f32_16x16x4_f32(false, a, false, bm,
                                                (short)0, acc, false, false);
  }
  const float bb = bias[ncol];
  float* Out = (which == 0) ? Q : (which == 1 ? Zr : V);
  const bool act = (which != 2);         // elu on q and z
  for (int r = 0; r < 8; ++r) {
    const int om = r + 8 * half;         // C/D layout: VGPR r -> M=r (lanes<16), M=8+r
    float v = acc[r] + bb;
    if (act) v = v > 0.f ? v : (__expf(v) - 1.f);
    Out[(mrow + om) * D + ncol] = v;
  }
}

// ---------------- K1b: normalize q rows ----------------
__global__ void norm_q_kernel(const float* __restrict__ Q, float* __restrict__ NQ) {
  const int row = blockIdx.x, tid = threadIdx.x;   // 64 threads (2 waves)
  __shared__ float r2[2];
  float q = Q[row * D + tid];
  float s = q * q;
  for (int off = 16; off > 0; off >>= 1) s += __shfl_down(s, off, 32);
  if ((tid & 31) == 0) r2[tid >> 5] = s;
  __syncthreads();
  const float inv = 1.f / fmaxf(sqrtf(r2[0] + r2[1]), 1e-8f);
  NQ[row * D + tid] = q * inv;
}

// ---------------- K0: per-batch prompt/history constants ----------------
__global__ void per_batch_prep(const int* __restrict__ user_id,
                               const float* __restrict__ P, const float* __restrict__ S,
                               const float* __restrict__ Wk, const float* __restrict__ bk,
                               const float* __restrict__ Wv, const float* __restrict__ bv,
                               const float* __restrict__ z_g, const float* __restrict__ z_b,
                               const float* __restrict__ s_g, const float* __restrict__ s_b,
                               const float* __restrict__ nam_g, const float* __restrict__ nam_b,
                               const float* __restrict__ prev_z_buf,
                               const float* __restrict__ prev_s_buf,
                               float* __restrict__ Pz, float* __restrict__ Ps,
                               float* __restrict__ Mb) {
  const int b = blockIdx.x, tid = threadIdx.x;     // 256 threads
  __shared__ float xb[LP * IN];
  __shared__ float zrow[LP * D], vrow[LP * D];
  __shared__ float buf[D * D];
  __shared__ float z64[D], pz64[D];
  __shared__ float red[16];
  for (int pass = 0; pass < 2; ++pass) {
    const float* Xp = (pass == 0 ? P : S) + b * LP * IN;
    for (int i = tid; i < LP * IN; i += 256) xb[i] = Xp[i];
    __syncthreads();
    // projections of the LP=8 rows (z = elu(xWk+bk), v = xWv+bv)
    for (int i = tid; i < LP * D; i += 256) {
      const int t = i >> 6, d = i & 63;
      float aK = bk[d], aV = bv[d];
      for (int k = 0; k < IN; ++k) {
        const float x = xb[t * IN + k];
        aK += x * Wk[k * D + d];
        aV += x * Wv[k * D + d];
      }
      zrow[i] = aK > 0.f ? aK : (__expf(aK) - 1.f);
      vrow[i] = aV;
    }
    __syncthreads();
    // totals over the LP steps (== last cumsum value)
    if (tid < D) {
      float s = 0.f;
      for (int t = 0; t < LP; ++t) s += zrow[t * D + tid];
      z64[tid] = s;
    }
    for (int j = 0; j < 16; ++j) {
      const int idx = tid + j * 256;
      const int d = idx >> 6, p = idx & 63;
      float s = 0.f;
      for (int t = 0; t < LP; ++t) s += zrow[t * D + d] * vrow[t * D + p];
      buf[idx] = s;
    }
    __syncthreads();
    // z LN (EPS shift cancels in centered term)
    const float zc = (tid < D) ? z64[tid] : 0.f;
    float s1, s2; block_red2_256(zc, zc * zc, red, s1, s2);
    const float mu = s1 / (float)D, var = s2 / (float)D - mu * mu;
    const float rstd = rsqrtf(var + LN_EPS);
    if (tid < D) pz64[tid] = (zc - mu) * rstd * z_g[tid] + z_b[tid];
    // s LN stats over 4096
    float ls = 0.f, lss = 0.f;
    for (int j = 0; j < 16; ++j) { const float x = buf[tid + j * 256]; ls += x; lss += x * x; }
    float S1, S2; block_red2_256(ls, lss, red, S1, S2);
    const float mus = S1 / 4096.f, vars = S2 / 4096.f - mus * mus;
    const float rstds = rsqrtf(vars + LN_EPS);
    if (pass == 0) {
      if (tid < D) Pz[b * D + tid] = pz64[tid];
      for (int j = 0; j < 16; ++j) {
        const int idx = tid + j * 256;
        Ps[b * D * D + idx] = (buf[idx] - mus) * rstds * s_g[idx] + s_b[idx];
      }
    } else {
      for (int j = 0; j < 16; ++j) {        // in-place LN (owner mapping, no race)
        const int idx = tid + j * 256;
        buf[idx] = (buf[idx] - mus) * rstds * s_g[idx] + s_b[idx];
      }
      __syncthreads();
      const int uid = user_id[b];
      float pzv = (tid < D) ? (prev_z_buf[uid * D + tid] + pz64[tid]) : 0.f;
      float n1, n2dummy; block_red2_256(pzv * pzv, 0.f, red, n1, n2dummy);
      const float inv = 1.f / (fmaxf(sqrtf(n1), 1e-8f) + EPSC);
      float ps_ = 0.f, pss = 0.f;
      for (int j = 0; j < 16; ++j) {
        const int idx = tid + j * 256;
        const float pn = (prev_s_buf[(size_t)uid * D * D + idx] + buf[idx]) * inv;
        ps_ += pn; pss += pn * pn;
      }
      float F1, F2; block_red2_256(ps_, pss, red, F1, F2);
      const float mun = F1 / 4096.f, varn = F2 / 4096.f - mun * mun;
      const float rstdn = rsqrtf(varn + LN_EPS);
      for (int j = 0; j < 16; ++j) {
        const int idx = tid + j * 256;
        const float pn = (prev_s_buf[(size_t)uid * D * D + idx] + buf[idx]) * inv;
        Mb[b * D * D + idx] = (pn - mun) * rstdn * nam_g[idx] + nam_b[idx];
      }
    }
    __syncthreads();
  }
}

// ---------------- K2a: per-chunk totals of z and outer(z,v) ----------------
__global__ void chunk_totals(const float* __restrict__ Zr, const float* __restrict__ V,
                             float* __restrict__ pZ, float* __restrict__ pS) {
  const int b = blockIdx.x >> 3, c = blockIdx.x & 7, tid = threadIdx.x;
  __shared__ float sS[D * D];
  __shared__ float sZ[D], zl[D], vl[D];
  for (int i = tid; i < D * D; i += 256) sS[i] = 0.f;
  if (tid < D) sZ[tid] = 0.f;
  __syncthreads();
  const int base = (b * L + c * CLEN) * D;
  for (int t = 0; t < CLEN; ++t) {
    if (tid < D) { zl[tid] = Zr[base + t * D + tid]; vl[tid] = V[base + t * D + tid]; }
    __syncthreads();
    if (tid < D) sZ[tid] += zl[tid];
    for (int j = 0; j < 16; ++j) {
      const int idx = tid + j * 256;
      sS[idx] += zl[idx >> 6] * vl[idx & 63];
    }
    __syncthreads();
  }
  if (tid < D) pZ[(b * CHUNKS + c) * D + tid] = sZ[tid];
  for (int j = 0; j < 16; ++j) {
    const int idx = tid + j * 256;
    pS[(b * CHUNKS + c) * D * D + idx] = sS[idx];
  }
}

// ---------------- K2b: exclusive prefix over chunks (in place) ----------------
__global__ void chunk_prefix(float* __restrict__ pZ, float* __restrict__ pS) {
  const int b = blockIdx.x, tid = threadIdx.x;
  if (tid < D) {
    float run = 0.f;
    for (int c = 0; c < CHUNKS; ++c) {
      const int idx = (b * CHUNKS + c) * D + tid;
      const float t = pZ[idx]; pZ[idx] = run; run += t;
    }
  }
  for (int j = 0; j < 16; ++j) {
    const int e = tid + j * 256;
    float run = 0.f;
    for (int c = 0; c < CHUNKS; ++c) {
      const int idx = (b * CHUNKS + c) * D * D + e;
      const float t = pS[idx]; pS[idx] = run; run += t;
    }
  }
}

// ---------------- K2c: main per-step scan ----------------
__global__ void scan_main(const float* __restrict__ Q, const float* __restrict__ Zr,
                          const float* __restrict__ V,
                          const float* __restrict__ pZ, const float* __restrict__ pS,
                          const float* __restrict__ Pz, const float* __restrict__ Ps,
                          const float* __restrict__ z_g, const float* __restrict__ z_b,
                          const float* __restrict__ s_g, const float* __restrict__ s_b,
                          const float* __restrict__ nam_g, const float* __restrict__ nam_b,
                          float* __restrict__ na_core, double* __restrict__ acc) {
  const int b = blockIdx.x >> 3, c = blockIdx.x & 7, tid = threadIdx.x;
  __shared__ float scum[D * D];     // running cumulative outer-product sum
  __shared__ float psb[D * D];      // prompt P_s last (LN'd), added per step
  __shared__ float zcum[D], zf[D], zl[D], vl[D], nq[D];
  __shared__ float red[16];
  __shared__ float napart[256];
  for (int j = 0; j < 16; ++j) {
    const int idx = tid + j * 256;
    scum[idx] = pS[(b * CHUNKS + c) * D * D + idx];
    psb[idx]  = Ps[b * D * D + idx];
  }
  if (tid < D) zcum[tid] = pZ[(b * CHUNKS + c) * D + tid];
  __syncthreads();
  float stdsum = 0.f, stdsq = 0.f;
  for (int t = 0; t < CLEN; ++t) {
    const int l = c * CLEN + t;
    const int row = (b * L + l) * D;
    if (tid < D) { zl[tid] = Zr[row + tid]; vl[tid] = V[row + tid]; }
    __syncthreads();
    if (tid < D) zcum[tid] += zl[tid];
    for (int j = 0; j < 16; ++j) {
      const int idx = tid + j * 256;
      scum[idx] += zl[idx >> 6] * vl[idx & 63];
    }
    // z LN (EPS cancels), add prompt z, l2 norm; q norm; std accumulation
    const float zc = (tid < D) ? zcum[tid] : 0.f;
    float s1, s2; block_red2_256(zc, zc * zc, red, s1, s2);
    const float muz = s1 / (float)D, varz = s2 / (float)D - muz * muz;
    const float rstdz = rsqrtf(varz + LN_EPS);
    if (tid < D) zf[tid] = (zc - muz) * rstdz * z_g[tid] + z_b[tid] + Pz[b * D + tid];
    __syncthreads();
    const float zfv = (tid < D) ? zf[tid] : 0.f;
    const float qv  = (tid < D) ? Q[row + tid] : 0.f;
    float zn2, qn2; block_red2_256(zfv * zfv, qv * qv, red, zn2, qn2);
    const float invz = 1.f / fmaxf(sqrtf(zn2), 1e-8f);
    const float invq = 1.f / fmaxf(sqrtf(qn2), 1e-8f);
    if (tid < D) {
      nq[tid] = qv * invq;
      const float r = qv / fmaxf(zfv, 1e-6f);
      stdsum += r; stdsq += r * r;
    }
    __syncthreads();
    // scum LN stats over 4096 (EPS cancels)
    float ls = 0.f, lss = 0.f;
    for (int j = 0; j < 16; ++j) { const float x = scum[tid + j * 256]; ls += x; lss += x * x; }
    float S1, S2; block_red2_256(ls, lss, red, S1, S2);
    const float mus = S1 / 4096.f, vars = S2 / 4096.f - mus * mus;
    const float rstds = rsqrtf(vars + LN_EPS);
    // s_full = LN(scum)+Ps stats; nam stats derived via invz scaling
    ls = 0.f; lss = 0.f;
    for (int j = 0; j < 16; ++j) {
      const int idx = tid + j * 256;
      const float sf = (scum[idx] - mus) * rstds * s_g[idx] + s_b[idx] + psb[idx];
      ls += sf; lss += sf * sf;
    }
    float F1, F2; block_red2_256(ls, lss, red, F1, F2);
    const float muf = F1 / 4096.f, varf = F2 / 4096.f - muf * muf;
    const float mean_n = invz * muf;                       // EPS cancels
    const float rstdn  = rsqrtf(invz * invz * varf + LN_EPS);
    // fused LN(nam)+contraction: column p, 16 d's per thread
    float part = 0.f;
    const int p = tid & 63, d0 = (tid >> 6) * 16;
    for (int d = d0; d < d0 + 16; ++d) {
      const int idx = d * D + p;
      const float sf = (scum[idx] - mus) * rstds * s_g[idx] + s_b[idx] + psb[idx];
      const float nm = (sf * invz - mean_n) * rstdn * nam_g[idx] + nam_b[idx];
      part += nq[d] * nm;
    }
    napart[tid] = part;
    __syncthreads();
    if (tid < D)
      na_core[row + tid] = napart[tid] + napart[tid + 64] +
                           napart[tid + 128] + napart[tid + 192];
    if (t + 1 < CLEN && tid < D) {
      __builtin_prefetch(&Zr[row + D + tid], 0, 0);   // global_prefetch_b8
      __builtin_prefetch(&V[row + D + tid], 0, 0);
    }
    __syncthreads();
  }
  float A1, A2; block_red2_256(stdsum, stdsq, red, A1, A2);
  if (tid == 0) { atomicAdd(&acc[0], (double)A1); atomicAdd(&acc[1], (double)A2); }
}

// ---------------- K3: history GEMM prev_na = NQ @ M_b via WMMA f32 ----------------
__global__ void prevna_wmma(const float* __restrict__ NQ, const float* __restrict__ Mb,
                            float* __restrict__ PrevNa) {
  const int lane = threadIdx.x & 31;
  const int wvid = threadIdx.x >> 5;
  const int half = lane >> 4;
  const int mn   = lane & 15;
  const int mrow = blockIdx.x * 16;
  const int b    = mrow / L;
  const float* W = Mb + b * D * D;
  const int ncol = wvid * 16 + mn;
  v8f acc = {};
  for (int kb = 0; kb < D; kb += 4) {
    const int k0 = kb + half * 2;
    v2f a, bm;
    a.x  = NQ[(mrow + mn) * D + k0];
    a.y  = NQ[(mrow + mn) * D + k0 + 1];
    bm.x = W[k0 * D + ncol];
    bm.y = W[(k0 + 1) * D + ncol];
    acc = __builtin_amdgcn_wmma_f32_16x16x4_f32(false, a, false, bm,
                                                (short)0, acc, false, false);
  }
  for (int r = 0; r < 8; ++r) {
    const int om = r + 8 * half;
    PrevNa[(mrow + om) * D + ncol] = acc[r];
  }
}

// ---------------- K4: combine + final LN over D ----------------
__global__ void combine_ln(const float* __restrict__ na_core, const float* __restrict__ prev_na,
                           const float* __restrict__ na_g, const float* __restrict__ na_b,
                           float* __restrict__ out) {
  const int row = blockIdx.x, tid = threadIdx.x;   // 64 threads
  __shared__ float r4[4];
  const float v = na_core[row * D + tid] + prev_na[row * D + tid];
  float s = v, ss = v * v;
  for (int off = 16; off > 0; off >>= 1) {
    s  += __shfl_down(s, off, 32);
    ss += __shfl_down(ss, off, 32);
  }
  if ((tid & 31) == 0) { r4[tid >> 5] = s; r4[2 + (tid >> 5)] = ss; }
  __syncthreads();
  const float mu  = (r4[0] + r4[1]) / (float)D;
  const float var = (r4[2] + r4[3]) / (float)D - mu * mu;
  out[row * D + tid] = (v - mu) * rsqrtf(var + LN_EPS) * na_g[tid] + na_b[tid];
}

// ---------------- init / finalize ----------------
__global__ void init_acc(double* acc) {
  if (threadIdx.x == 0) { acc[0] = 0.0; acc[1] = 0.0; }
}
__global__ void finalize_std(const double* __restrict__ acc, float* __restrict__ out) {
  if (threadIdx.x == 0) {
    const double N = (double)(ROWS * D);
    const double mean = acc[0] / N;
    const double var  = (acc[1] - N * mean * mean) / (N - 1.0);
    out[ROWS * D] = (float)sqrt(var > 0.0 ? var : 0.0);
  }
}

extern "C" void kernel_launch(void* const* d_in, const int* in_sizes, int n_in,
                              void* d_out, int out_size, void* d_ws, size_t ws_size,
                              hipStream_t stream) {
  const int*   user_id    = (const int*)  d_in[0];
  const float* seqs_emb   = (const float*)d_in[1];
  const float* P          = (const float*)d_in[2];
  const float* S          = (const float*)d_in[3];
  const float* Wq         = (const float*)d_in[4];
  const float* bq         = (const float*)d_in[5];
  const float* Wk         = (const float*)d_in[6];
  const float* bk         = (const float*)d_in[7];
  const float* Wv         = (const float*)d_in[8];
  const float* bv         = (const float*)d_in[9];
  const float* z_g        = (const float*)d_in[10];
  const float* z_b        = (const float*)d_in[11];
  const float* s_g        = (const float*)d_in[12];
  const float* s_b        = (const float*)d_in[13];
  const float* nam_g      = (const float*)d_in[14];
  const float* nam_b      = (const float*)d_in[15];
  const float* na_g       = (const float*)d_in[16];
  const float* na_b       = (const float*)d_in[17];
  const float* prev_z_buf = (const float*)d_in[18];
  const float* prev_s_buf = (const float*)d_in[19];
  float* out = (float*)d_out;

  // workspace layout (floats); total ~15.3 MB
  float* ws     = (float*)d_ws;
  float* Qb     = ws;                         // ROWS*D
  float* Zr     = Qb + (size_t)ROWS * D;      // ROWS*D
  float* Vb     = Zr + (size_t)ROWS * D;      // ROWS*D
  float* NQ     = Vb + (size_t)ROWS * D;      // ROWS*D
  float* NaCore = NQ + (size_t)ROWS * D;      // ROWS*D
  float* PrevNa = NaCore + (size_t)ROWS * D;  // ROWS*D
  float* Pz     = PrevNa + (size_t)ROWS * D;  // B*D
  float* Ps     = Pz + B * D;                 // B*D*D
  float* Mb     = Ps + B * D * D;             // B*D*D
  float* pZ     = Mb + B * D * D;             // B*CHUNKS*D
  float* pS     = pZ + B * CHUNKS * D;        // B*CHUNKS*D*D
  double* acc   = (double*)(pS + (size_t)B * CHUNKS * D * D);  // 2 doubles, 8B aligned

  init_acc<<<1, 32, 0, stream>>>(acc);
  per_batch_prep<<<B, 256, 0, stream>>>(user_id, P, S, Wk, bk, Wv, bv,
                                        z_g, z_b, s_g, s_b, nam_g, nam_b,
                                        prev_z_buf, prev_s_buf, Pz, Ps, Mb);
  proj_wmma<<<dim3(ROWS / 16, 3, 1), 128, 0, stream>>>(seqs_emb, Wq, bq, Wk, bk, Wv, bv,
                                                       Qb, Zr, Vb);
  norm_q_kernel<<<ROWS, 64, 0, stream>>>(Qb, NQ);
  chunk_totals<<<B * CHUNKS, 256, 0, stream>>>(Zr, Vb, pZ, pS);
  chunk_prefix<<<B, 256, 0, stream>>>(pZ, pS);
  scan_main<<<B * CHUNKS, 256, 0, stream>>>(Qb, Zr, Vb, pZ, pS, Pz, Ps,
                                            z_g, z_b, s_g, s_b, nam_g, nam_b,
                                            NaCore, acc);
  prevna_wmma<<<ROWS / 16, 128, 0, stream>>>(NQ, Mb, PrevNa);
  combine_ln<<<ROWS, 64, 0, stream>>>(NaCore, PrevNa, na_g, na_b, out);
  finalize_std<<<1, 32, 0, stream>>>(acc, out);
}